// BU_Net_Loss_111669149714
// MI455X (gfx1250) — compile-verified
//
#include <hip/hip_runtime.h>

// ---------------------------------------------------------------------------
// BU-Net loss (weighted CE + dice) for pred[16,4,512,512] f32, target int.
//
// MI455X reasoning: memory-bound reduction. Traffic budget:
//   pass1: read target 16 MiB, write compact u8 labels 4 MiB (+4-bin histogram)
//   pass2: read pred 64 MiB + compact labels 4 MiB, fused softmax+all sums
//   pass3: tiny finalize; cross-block sums done with V_WMMA_F32_16X16X4_F32
// ~88 MiB total -> ~3.8 us at 23.3 TB/s. Compute (21M exp/log) is far under
// the trans-op ceiling, so wide b128 loads + one fused pass is optimal.
// ---------------------------------------------------------------------------

#define NN 16
#define CC 4
#define HH 512
#define WWD 512
#define HWC (HH * WWD)            // 262144 pixels per image
#define CHW (CC * HWC)
#define NHW (NN * HWC)            // 4194304 pixels total
#define TPB 256
#define BLOCKS_PER_N 128
#define PIX_PER_BLOCK (HWC / BLOCKS_PER_N)   // 2048
#define NQ 11                     // quantities: w, lw, tw, inter[4], union[4]
#define EPSV 1e-6f

typedef __attribute__((ext_vector_type(2))) float v2f;
typedef __attribute__((ext_vector_type(8))) float v8f;

__global__ void k0_init(unsigned int* counts) {
    if (threadIdx.x < CC) counts[threadIdx.x] = 0u;
}

// Pass 1: global class histogram + compact labels to u8 (saves 12 MiB on pass2)
__global__ __launch_bounds__(TPB) void k1_hist_compact(
        const int* __restrict__ tgt, unsigned char* __restrict__ ct,
        unsigned int* __restrict__ counts) {
    __shared__ unsigned int h[CC];
    const int tid = threadIdx.x;
    if (tid < CC) h[tid] = 0u;
    __syncthreads();

    const int gi = blockIdx.x * TPB + tid;          // int4 group, exact cover
    const int4 tv = ((const int4*)tgt)[gi];
    const int t0 = min(max(tv.x, 0), CC - 1);
    const int t1 = min(max(tv.y, 0), CC - 1);
    const int t2 = min(max(tv.z, 0), CC - 1);
    const int t3 = min(max(tv.w, 0), CC - 1);

    uchar4 pk;
    pk.x = (unsigned char)t0; pk.y = (unsigned char)t1;
    pk.z = (unsigned char)t2; pk.w = (unsigned char)t3;
    ((uchar4*)ct)[gi] = pk;

    unsigned int cnt[CC] = {0u, 0u, 0u, 0u};
    cnt[t0]++; cnt[t1]++; cnt[t2]++; cnt[t3]++;
#pragma unroll
    for (int j = 0; j < CC; ++j) {
        int v = (int)cnt[j];
        for (int off = 16; off; off >>= 1) v += __shfl_xor(v, off);
        if ((tid & 31) == 0) atomicAdd(&h[j], (unsigned int)v);
    }
    __syncthreads();
    if (tid < CC) atomicAdd(&counts[tid], h[tid]);   // integer atomics: deterministic
}

// Pass 2: fused log-softmax + all weighted reductions, one block = slice of one n.
// Deterministic: each block writes its own partial slot (no float atomics).
__global__ __launch_bounds__(TPB) void k2_main(
        const float* __restrict__ pred, const unsigned char* __restrict__ ct,
        const unsigned int* __restrict__ counts, float* __restrict__ partials) {
    __shared__ float cw_s[CC];
    __shared__ float wred[TPB / 32][NQ];
    const int tid = threadIdx.x;
    if (tid < CC) cw_s[tid] = 1.0f / ((float)counts[tid] + EPSV);
    __syncthreads();
    const float cwt[CC] = {cw_s[0], cw_s[1], cw_s[2], cw_s[3]};

    const int n   = blockIdx.x / BLOCKS_PER_N;
    const int blk = blockIdx.x % BLOCKS_PER_N;
    const float* __restrict__ p0        = pred + (size_t)n * CHW;
    const unsigned char* __restrict__ tb = ct  + (size_t)n * HWC;

    float a_w = 0.f, a_lw = 0.f, a_tw = 0.f;
    float a_in[CC] = {0.f, 0.f, 0.f, 0.f};
    float a_un[CC] = {0.f, 0.f, 0.f, 0.f};

    const int qbase = blk * PIX_PER_BLOCK + tid * 4;
    // gfx1250 global_prefetch_b8 for the second tile of each channel plane
    __builtin_prefetch(p0 + 0 * HWC + qbase + TPB * 4, 0, 1);
    __builtin_prefetch(p0 + 1 * HWC + qbase + TPB * 4, 0, 1);
    __builtin_prefetch(p0 + 2 * HWC + qbase + TPB * 4, 0, 1);
    __builtin_prefetch(p0 + 3 * HWC + qbase + TPB * 4, 0, 1);

#pragma unroll
    for (int it = 0; it < PIX_PER_BLOCK / (TPB * 4); ++it) {
        const int q = qbase + it * (TPB * 4);
        const float4 x0 = *(const float4*)(p0 + 0 * HWC + q);   // b128 loads
        const float4 x1 = *(const float4*)(p0 + 1 * HWC + q);
        const float4 x2 = *(const float4*)(p0 + 2 * HWC + q);
        const float4 x3 = *(const float4*)(p0 + 3 * HWC + q);
        const uchar4 tq = *(const uchar4*)(tb + q);
#pragma unroll
        for (int j = 0; j < 4; ++j) {
            const float e0 = ((const float*)&x0)[j];
            const float e1 = ((const float*)&x1)[j];
            const float e2 = ((const float*)&x2)[j];
            const float e3 = ((const float*)&x3)[j];
            const int   t  = (int)((const unsigned char*)&tq)[j];
            const float m  = fmaxf(fmaxf(e0, e1), fmaxf(e2, e3));
            const float s  = __expf(e0 - m) + __expf(e1 - m) +
                             __expf(e2 - m) + __expf(e3 - m);
            const float lse = __logf(s) + m;
            const float w   = cwt[t];
            const float xt  = (t == 0) ? e0 : (t == 1) ? e1 : (t == 2) ? e2 : e3;
            a_lw = fmaf(xt - lse, w, a_lw);
            a_w += w;
            const float tw = (float)t * w;
            a_tw += tw;
            a_in[0] = fmaf(e0, tw, a_in[0]); a_un[0] = fmaf(e0, w, a_un[0]);
            a_in[1] = fmaf(e1, tw, a_in[1]); a_un[1] = fmaf(e1, w, a_un[1]);
            a_in[2] = fmaf(e2, tw, a_in[2]); a_un[2] = fmaf(e2, w, a_un[2]);
            a_in[3] = fmaf(e3, tw, a_in[3]); a_un[3] = fmaf(e3, w, a_un[3]);
        }
    }

    float v[NQ] = {a_w, a_lw, a_tw,
                   a_in[0], a_in[1], a_in[2], a_in[3],
                   a_un[0], a_un[1], a_un[2], a_un[3]};
#pragma unroll
    for (int j = 0; j < NQ; ++j) {
        float x = v[j];
        for (int off = 16; off; off >>= 1) x += __shfl_xor(x, off);
        v[j] = x;
    }
    if ((tid & 31) == 0) {
        const int wid = tid >> 5;
#pragma unroll
        for (int j = 0; j < NQ; ++j) wred[wid][j] = v[j];
    }
    __syncthreads();
    if (tid < NQ) {
        float s = 0.f;
#pragma unroll
        for (int wv = 0; wv < TPB / 32; ++wv) s += wred[wv][tid];
        // layout: partials[(q*NN + n)*BLOCKS_PER_N + blk] -> contiguous per (q,n)
        partials[((size_t)tid * NN + n) * BLOCKS_PER_N + blk] = s;
    }
}

// Pass 3 (1 wave): sum 128 block-partials per (quantity,n) using
// V_WMMA_F32_16X16X4_F32 with A = ones. Every D element is a column-sum of B,
// so sum(all D)/16 == sum(all B) regardless of B's lane layout -> exact f32.
__global__ __launch_bounds__(32) void k3_final(
        const float* __restrict__ partials, float* __restrict__ out) {
    __shared__ float totals[NQ * NN];
    const int lane = threadIdx.x;
    for (int qn = 0; qn < NQ * NN; ++qn) {
        const float* P = partials + (size_t)qn * BLOCKS_PER_N;
        float s;
#if __has_builtin(__builtin_amdgcn_wmma_f32_16x16x4_f32)
        v8f d = {0.f, 0.f, 0.f, 0.f, 0.f, 0.f, 0.f, 0.f};
        v2f a; a.x = 1.f; a.y = 1.f;          // A = ones(16x4)
#pragma unroll
        for (int ch = 0; ch < 2; ++ch) {      // 2 x 64 values fill B per chunk
            v2f b;
            b.x = P[ch * 64 + 2 * lane + 0];
            b.y = P[ch * 64 + 2 * lane + 1];
            d = __builtin_amdgcn_wmma_f32_16x16x4_f32(
                    false, a, false, b, (short)0, d, false, false);
        }
        s = (d[0] + d[1] + d[2] + d[3] + d[4] + d[5] + d[6] + d[7]) * (1.0f / 16.0f);
#else
        s = P[2 * lane] + P[2 * lane + 1] + P[64 + 2 * lane] + P[65 + 2 * lane];
#endif
        for (int off = 16; off; off >>= 1) s += __shfl_xor(s, off);
        if (lane == 0) totals[qn] = s;
    }
    __syncthreads();
    if (lane == 0) {
        float wce = 0.f;
        for (int n = 0; n < NN; ++n)
            wce += totals[1 * NN + n] / totals[0 * NN + n];
        wce = -wce / (float)NN;
        float dice = 0.f;
        for (int n = 0; n < NN; ++n) {
            const float twsum = totals[2 * NN + n];
            for (int c = 0; c < CC; ++c) {
                const float inter = totals[(3 + c) * NN + n];
                const float uni   = totals[(7 + c) * NN + n] + twsum;
                dice += 1.f - (2.f * inter + 1.f) / (uni + 1.f);   // SMOOTH=1
            }
        }
        dice /= (float)(NN * CC);
        out[0] = wce + dice;
    }
}

extern "C" void kernel_launch(void* const* d_in, const int* in_sizes, int n_in,
                              void* d_out, int out_size, void* d_ws, size_t ws_size,
                              hipStream_t stream) {
    const float* pred = (const float*)d_in[0];
    const int*   tgt  = (const int*)d_in[1];

    // workspace layout: [0, NHW) u8 compact labels | 4 u32 counts | partials f32
    unsigned char* ct      = (unsigned char*)d_ws;
    unsigned int*  counts  = (unsigned int*)((char*)d_ws + (size_t)NHW);
    float*         partial = (float*)((char*)d_ws + (size_t)NHW + 16);
    float*         out     = (float*)d_out;

    k0_init<<<1, 32, 0, stream>>>(counts);
    k1_hist_compact<<<NHW / 4 / TPB, TPB, 0, stream>>>(tgt, ct, counts);
    k2_main<<<NN * BLOCKS_PER_N, TPB, 0, stream>>>(pred, ct, counts, partial);
    k3_final<<<1, 32, 0, stream>>>(partial, out);
}